// PFBAnalysisLayer_8924942041335
// MI455X (gfx1250) — compile-verified
//
#include <hip/hip_runtime.h>
#include <hip/hip_bf16.h>

typedef __attribute__((ext_vector_type(2))) float v2f;
typedef __attribute__((ext_vector_type(4))) float v4f;
typedef __attribute__((ext_vector_type(8))) float v8f;

#define FFT_LEN   256
#define WIN_LEN   768
#define FRM_LEN   128
#define PPF_TAP   3
#define N_SAMP    262144
#define T_FRAMES  2048          // N_SAMP / FRM_LEN
#define N_BC      32            // 8 * 4
#define BLK_FR    32            // frames per block
#define FBLKS     (T_FRAMES / BLK_FR)   // 64
#define NTHREADS  288           // 9 waves of 32
#define SP        260           // LDS row pitch (floats): 260 % 64 banks = 4 -> conflict-free B frags
#define STEP      (6.283185307179586f / 256.0f)

// Fused PFB window + DFT-as-WMMA kernel.
// Stage A: S[f][j] = sum_p x[128*(t0+f) - 640 + p*256 + j] * g[p*256+j], g = reversed lpf.
// Stage B: spec[k] = sum_j S[j] * exp(-2*pi*i*k*(255-j)/256), via V_WMMA_F32_16X16X4_F32
//          with A = twiddles (generated by per-lane phasor recurrence), B = S from LDS.
__global__ __launch_bounds__(NTHREADS)
void pfb_analysis_wmma_kernel(const float* __restrict__ x,
                              const float* __restrict__ lpf,
                              float2* __restrict__ out)
{
    __shared__ float g_lds[WIN_LEN];
    __shared__ float s_lds[BLK_FR * SP];

    const int bc = blockIdx.x >> 6;         // 0..31  (b*4 + c)
    const int fb = blockIdx.x & 63;         // frame block 0..63
    const int t0 = fb * BLK_FR;
    const float* __restrict__ xg = x + (size_t)bc * N_SAMP;

    // ---- load reversed filter into LDS ----
    for (int i = threadIdx.x; i < WIN_LEN; i += NTHREADS)
        g_lds[i] = lpf[WIN_LEN - 1 - i];
    __syncthreads();

    // ---- Stage A: windowed polyphase sum -> S in LDS (vectorized by 4 along j) ----
    for (int e = threadIdx.x; e < BLK_FR * (FFT_LEN / 4); e += NTHREADS) {
        const int f  = e >> 6;              // frame within block
        const int j4 = (e & 63) << 2;       // j = j4..j4+3
        const int base = FRM_LEN * (t0 + f) - (WIN_LEN - FRM_LEN) + j4;
        v4f acc = {0.f, 0.f, 0.f, 0.f};
#pragma unroll
        for (int p = 0; p < PPF_TAP; ++p) {
            const int idx = base + p * FFT_LEN;
            const v4f gv = *(const v4f*)(g_lds + p * FFT_LEN + j4);
            // idx is a multiple of 4: window is either fully valid or fully zero-padded
            if (idx >= 0) {
                const v4f xv = *(const v4f*)(xg + idx);
                acc += xv * gv;
            }
        }
        *(v4f*)(s_lds + f * SP + j4) = acc;
    }
    __syncthreads();

    // ---- Stage B: DFT via WMMA fp32 16x16x4 ----
    const int wave = threadIdx.x >> 5;      // 0..8 -> k-tile
    const int lane = threadIdx.x & 31;
    const int l16  = lane & 15;
    const int hi   = lane >> 4;             // 0: K slots {0,1}, 1: K slots {2,3}
    const int kcol = wave * 16 + l16;       // twiddle row k for this lane (0..143)

    // phasors for the two K slots of the A fragment; theta(j) = 2*pi*((k*(255-j)) mod 256)/256
    float cx, sx, cy, sy, cd, sd;
    {
        const int jx = 2 * hi, jy = jx + 1;
        const int mx = (kcol * (255 - jx)) & 255;
        const int my = (kcol * (255 - jy)) & 255;
        __sincosf((float)mx * STEP, &sx, &cx);
        __sincosf((float)my * STEP, &sy, &cy);
        const int md = (kcol * 4) & 255;    // per-step rotation: j += 4 -> angle -= 2*pi*md/256
        __sincosf(-(float)md * STEP, &sd, &cd);
    }

    v8f ar[2] = {};                         // real accumulators, frame-tiles 0/1
    v8f ai[2] = {};                         // imag accumulators
    const float* sp = s_lds + l16 * SP + 2 * hi;

#pragma unroll 4
    for (int kk = 0; kk < FFT_LEN / 4; ++kk) {
        const v2f a_re = {cx, cy};          // cos(theta)
        const v2f a_im = {-sx, -sy};        // -sin(theta)
#pragma unroll
        for (int ft = 0; ft < 2; ++ft) {
            const v2f b = *(const v2f*)(sp + ft * 16 * SP);   // ds_load_b64, conflict-free
            ar[ft] = __builtin_amdgcn_wmma_f32_16x16x4_f32(
                false, a_re, false, b, (short)0, ar[ft], false, false);
            ai[ft] = __builtin_amdgcn_wmma_f32_16x16x4_f32(
                false, a_im, false, b, (short)0, ai[ft], false, false);
        }
        sp += 4;
        // rotate both phasors by e^{-i*2*pi*(4k mod 256)/256}
        const float ncx = cx * cd - sx * sd; sx = sx * cd + cx * sd; cx = ncx;
        const float ncy = cy * cd - sy * sd; sy = sy * cd + cy * sd; cy = ncy;
    }

    // ---- store: D lane l holds (k_local = r + 8*hi, t_local = l16) -> coalesced float2 ----
#pragma unroll
    for (int ft = 0; ft < 2; ++ft) {
        const int t = t0 + 16 * ft + l16;
#pragma unroll
        for (int r = 0; r < 8; ++r) {
            const int k = wave * 16 + r + 8 * hi;
            if (k < FFT_LEN / 2 + 1) {
                float2 v; v.x = ar[ft][r]; v.y = ai[ft][r];
                out[(size_t)(bc * (FFT_LEN / 2 + 1) + k) * T_FRAMES + t] = v;
            }
        }
    }
}

extern "C" void kernel_launch(void* const* d_in, const int* in_sizes, int n_in,
                              void* d_out, int out_size, void* d_ws, size_t ws_size,
                              hipStream_t stream)
{
    (void)in_sizes; (void)n_in; (void)d_ws; (void)ws_size; (void)out_size;
    const float* x   = (const float*)d_in[0];   // (8,4,262144) f32
    const float* lpf = (const float*)d_in[1];   // (768,) f32
    float2* out = (float2*)d_out;               // (8,4,129,2048) complex64

    dim3 grid(N_BC * FBLKS);                    // 2048 blocks
    dim3 block(NTHREADS);                       // 9 wave32
    pfb_analysis_wmma_kernel<<<grid, block, 0, stream>>>(x, lpf, out);
}